// TaylorSelfAttentionModule_58548994179564
// MI455X (gfx1250) — compile-verified
//
#include <hip/hip_runtime.h>

typedef __attribute__((ext_vector_type(2))) float v2f;
typedef __attribute__((ext_vector_type(4))) float v4f;
typedef __attribute__((ext_vector_type(8))) float v8f;

#define TSA_BM 64
#define TSA_BN 128
#define TSA_BK 32
#define TSA_LDSS 36   // padded LDS row stride (floats): conflict-free b64 reads, 16B-aligned rows

// C[M,N] = A(MxK) * B(KxN)  (B_IS_NT=false, B row-major [K x N])
//        = A(MxK) * Bt^T    (B_IS_NT=true,  Bt row-major [N x K])
// Block tile 64x128, 256 threads = 8 wave32s.
// Wave w: row-tile tr = w&3, col-tiles tc0..tc0+3 with tc0 = (w>>2)*4.
// Double-buffered LDS with register staging: global loads of tile k+1 overlap
// the WMMA stream on tile k; one workgroup barrier per K-tile.
template <bool B_IS_NT>
__global__ __launch_bounds__(256) void tsa_gemm_wmma_f32(
    const float* __restrict__ Ag, const float* __restrict__ Bg,
    float* __restrict__ Cg,
    int K, int lda, int ldb, int ldc,
    long strideA, long strideB, long strideC,
    const float* __restrict__ bias)
{
    __shared__ float Alds[2][TSA_BM][TSA_LDSS];
    __shared__ float Blds[2][TSA_BN][TSA_LDSS];   // always stored [n][k]

    const float* A = Ag + (long)blockIdx.z * strideA;
    const float* B = Bg + (long)blockIdx.z * strideB;
    float*       C = Cg + (long)blockIdx.z * strideC;

    const int tid  = threadIdx.x;
    const int lane = tid & 31;
    const int wave = tid >> 5;          // 0..7
    const int tr   = wave & 3;          // 16-row tile within block
    const int tc0  = (wave >> 2) << 2;  // 0 or 4: four 16-col tiles

    const int m0 = blockIdx.y * TSA_BM;
    const int n0 = blockIdx.x * TSA_BN;

    v8f acc[4];
    #pragma unroll
    for (int j = 0; j < 4; ++j) acc[j] = (v8f){0.f,0.f,0.f,0.f,0.f,0.f,0.f,0.f};

    v4f aReg[2];   // 64x32 A tile: 2 float4 per thread
    v4f bReg[4];   // 128x32 (NT) or 32x128 (NN) B tile: 4 float4 per thread

    // ---- global -> registers for K-tile starting at k0 ----
    auto loadTile = [&](int k0) {
        #pragma unroll
        for (int r = 0; r < 2; ++r) {
            int e   = tid + r * 256;       // 0..511
            int row = e >> 3;              // 64 rows
            int c4  = (e & 7) << 2;        // 8 float4 per row
            aReg[r] = *(const v4f*)(A + (long)(m0 + row) * lda + (k0 + c4));
        }
        if (B_IS_NT) {
            #pragma unroll
            for (int r = 0; r < 4; ++r) {
                int e   = tid + r * 256;   // 0..1023
                int row = e >> 3;          // 128 n-rows
                int c4  = (e & 7) << 2;    // k
                bReg[r] = *(const v4f*)(B + (long)(n0 + row) * ldb + (k0 + c4));
            }
        } else {
            #pragma unroll
            for (int r = 0; r < 4; ++r) {
                int e  = tid + r * 256;
                int kr = e >> 5;           // 32 k-rows
                int cg = (e & 31) << 2;    // 32 float4 across n
                bReg[r] = *(const v4f*)(B + (long)(k0 + kr) * ldb + (n0 + cg));
            }
        }
    };

    // ---- registers -> LDS buffer ----
    auto stashTile = [&](int buf) {
        #pragma unroll
        for (int r = 0; r < 2; ++r) {
            int e   = tid + r * 256;
            int row = e >> 3;
            int c4  = (e & 7) << 2;
            *(v4f*)&Alds[buf][row][c4] = aReg[r];
        }
        if (B_IS_NT) {
            #pragma unroll
            for (int r = 0; r < 4; ++r) {
                int e   = tid + r * 256;
                int row = e >> 3;
                int c4  = (e & 7) << 2;
                *(v4f*)&Blds[buf][row][c4] = bReg[r];
            }
        } else {
            #pragma unroll
            for (int r = 0; r < 4; ++r) {
                int e  = tid + r * 256;
                int kr = e >> 5;
                int cg = (e & 31) << 2;
                Blds[buf][cg + 0][kr] = bReg[r].x;   // transpose into [n][k]
                Blds[buf][cg + 1][kr] = bReg[r].y;
                Blds[buf][cg + 2][kr] = bReg[r].z;
                Blds[buf][cg + 3][kr] = bReg[r].w;
            }
        }
    };

    const int fm  = tr  * 16 + (lane & 15);  // A row (lanes 0-15 == 16-31)
    const int fn0 = tc0 * 16 + (lane & 15);  // first B col

    // ---- WMMA over one staged K-tile ----
    auto compute = [&](int buf) {
        #pragma unroll
        for (int kk = 0; kk < TSA_BK; kk += 4) {
            // 16x4 f32 A layout: VGPR0/1 = K {0,1} (lanes 0-15), K {2,3} (lanes 16-31)
            int kb = kk + ((lane >> 4) << 1);
            v2f a = *(const v2f*)&Alds[buf][fm][kb];
            #pragma unroll
            for (int j = 0; j < 4; ++j) {
                v2f b = *(const v2f*)&Blds[buf][fn0 + 16 * j][kb];
                acc[j] = __builtin_amdgcn_wmma_f32_16x16x4_f32(
                    false, a, false, b, (short)0, acc[j], false, false);
            }
        }
    };

    // ---- pipelined main loop ----
    const int nK = K / TSA_BK;
    loadTile(0);
    stashTile(0);
    __syncthreads();
    for (int kt = 0; kt < nK; ++kt) {
        int cur = kt & 1;
        if (kt + 1 < nK) loadTile((kt + 1) * TSA_BK);  // HBM loads in flight...
        compute(cur);                                   // ...behind the WMMA stream
        if (kt + 1 < nK) stashTile(cur ^ 1);
        __syncthreads();
    }

    // ---- store: 16x16 f32 C/D layout: VGPR i = row Mbase+i, lanes 16-31 -> rows +8 ----
    const int cn = n0 + tc0 * 16 + (lane & 15);
    const int cm = m0 + tr * 16 + ((lane >> 4) << 3);
    #pragma unroll
    for (int j = 0; j < 4; ++j) {
        float bi = (bias != nullptr) ? bias[cn + 16 * j] : 0.f;
        #pragma unroll
        for (int i = 0; i < 8; ++i) {
            C[(long)(cm + i) * ldc + (cn + 16 * j)] = acc[j][i] + bi;
        }
    }
}

// b_fused[o] = b_out[o] + sum_f W_out[o,f] * b_head_flat[f]   (512 outputs, trivial)
__global__ __launch_bounds__(256) void tsa_bias_fuse(
    const float* __restrict__ Wo, const float* __restrict__ bh,
    const float* __restrict__ bo, float* __restrict__ bf)
{
    int o = blockIdx.x * blockDim.x + threadIdx.x;
    if (o >= 512) return;
    const float* w = Wo + (long)o * 1024;
    float s = bo[o];
    for (int f = 0; f < 1024; ++f) s = fmaf(w[f], bh[f], s);
    bf[o] = s;
}

extern "C" void kernel_launch(void* const* d_in, const int* in_sizes, int n_in,
                              void* d_out, int out_size, void* d_ws, size_t ws_size,
                              hipStream_t stream) {
    (void)in_sizes; (void)n_in; (void)out_size; (void)ws_size;
    // Dims: B=16, L=1024, D=512, F=H*DM=1024, O=512
    const float* x  = (const float*)d_in[0];  // [16,1024,512]
    const float* Wh = (const float*)d_in[1];  // [16,64,1024] == Wc [1024(f) x 1024(m)]
    const float* bh = (const float*)d_in[2];  // [1024]
    const float* Wo = (const float*)d_in[3];  // [512,1024]
    const float* bo = (const float*)d_in[4];  // [512]
    float* out = (float*)d_out;               // [16,1024,512]

    // Workspace layout (floats): Wfused[512*1024] | b_fused[512] | U[16*512*512]
    float* Wf = (float*)d_ws;
    float* bf = Wf + 512 * 1024;
    float* U  = bf + 512;

    // (0) fold biases: b_fused = b_out + W_out @ b_head_flat
    tsa_bias_fuse<<<dim3(2), dim3(256), 0, stream>>>(Wo, bh, bo, bf);

    // (1) Wfused [512 x 1024] = W_out [512 x 1024] @ Wc [1024 x 1024]   (NN)
    tsa_gemm_wmma_f32<false><<<dim3(8, 8, 1), dim3(256), 0, stream>>>(
        Wo, Wh, Wf, /*K=*/1024, /*lda=*/1024, /*ldb=*/1024, /*ldc=*/1024,
        0L, 0L, 0L, nullptr);

    // (2) per batch: U[b] [512 x 512] = Wfused [512 x 1024] @ x[b] [1024 x 512]   (NN)
    tsa_gemm_wmma_f32<false><<<dim3(4, 8, 16), dim3(256), 0, stream>>>(
        Wf, x, U, /*K=*/1024, /*lda=*/1024, /*ldb=*/512, /*ldc=*/512,
        0L, (long)1024 * 512, (long)512 * 512, nullptr);

    // (3) per batch: out[b] [1024 x 512] = x[b] [1024 x 512] @ U[b]^T + b_fused   (NT)
    tsa_gemm_wmma_f32<true><<<dim3(4, 16, 16), dim3(256), 0, stream>>>(
        x, U, out, /*K=*/512, /*lda=*/512, /*ldb=*/512, /*ldc=*/512,
        (long)1024 * 512, (long)512 * 512, (long)1024 * 512, bf);
}